// HungarianMatcher_412316860439
// MI455X (gfx1250) — compile-verified
//
#include <hip/hip_runtime.h>
#include <hip/hip_bf16.h>

// ---------------------------------------------------------------------------
// DETR HungarianMatcher cost matrix for MI455X (gfx1250, wave32).
//
//   C[n,t] = 5*L1(pbox[n], tbox[t]) + 2*(pos-neg)[n, id[t]] - 2*GIoU(n,t)
//
// Kernel 1: build class-cost table cc[n][c] = 2*(pos-neg), n<14400, c<92
//           (c==91 zero-padded so WMMA K covers 0..91 in 23 steps of 4).
// Kernel 2: tiled [N,T] kernel. Class gather done as f32 WMMA against a
//           one-hot B matrix (exact in f32), L1+GIoU on VALU, NT stores.
// ---------------------------------------------------------------------------

#define NQ      14400      // 16*900 predictions
#define NCLS    91
#define KPAD    92         // 23 * 4
#define NT      1600       // targets
#define COSTW_BBOX 5.0f
#define COSTW_GIOU 2.0f

typedef float v2f __attribute__((ext_vector_type(2)));
typedef float v8f __attribute__((ext_vector_type(8)));

// ------------------------- Kernel 1: class-cost table -----------------------
__global__ void __launch_bounds__(256)
hm_class_table(const float* __restrict__ logits, float* __restrict__ cc)
{
    int idx = blockIdx.x * 256 + threadIdx.x;
    if (idx >= NQ * KPAD) return;
    int n = idx / KPAD;
    int c = idx - n * KPAD;
    float v = 0.0f;
    if (c < NCLS) {
        float x = logits[n * NCLS + c];
        float p = 1.0f / (1.0f + __expf(-x));          // sigmoid
        float om = 1.0f - p;
        float pos = 0.25f * om * om * (-logf(p  + 1e-8f));
        float neg = 0.75f * p  * p  * (-logf(om + 1e-8f));
        v = 2.0f * (pos - neg);                        // fold COST_CLASS=2
    }
    cc[idx] = v;
}

// ------------------------- Kernel 2: cost matrix ----------------------------
// block: 320 threads = 10 waves; tile = 16 rows x 160 cols (10 WMMA tiles)
// grid:  (NT/160, NQ/16) = (10, 900)
__global__ void __launch_bounds__(320)
hm_cost(const float* __restrict__ cc,        // [NQ][KPAD] class table (x2)
        const float* __restrict__ pred_box,  // [NQ][4] cxcywh
        const float* __restrict__ tgt_box,   // [NT][4] cxcywh
        const int*   __restrict__ tgt_id,    // [NT]
        float* __restrict__ out)             // [NQ][NT]
{
    __shared__ float ccs[16 * KPAD];   // class-cost tile, row stride 92 (bank-safe)
    __shared__ float pbs[16 * 9];      // per row: cx,cy,w,h, x0,y0,x1,y1, area

    const int tid     = threadIdx.x;
    const int rowBase = blockIdx.y * 16;

    // cooperative fill of the class-cost tile (16 x 92 f32 = 5888 B)
    for (int i = tid; i < 16 * KPAD; i += 320)
        ccs[i] = cc[(rowBase + (i / KPAD)) * KPAD + (i % KPAD)];

    // pred-box preprocessing for the 16 rows
    if (tid < 16) {
        const float* pb = pred_box + (rowBase + tid) * 4;
        float cx = pb[0], cy = pb[1], w = pb[2], h = pb[3];
        float* s = pbs + tid * 9;
        s[0] = cx; s[1] = cy; s[2] = w; s[3] = h;
        s[4] = cx - 0.5f * w;  s[5] = cy - 0.5f * h;
        s[6] = cx + 0.5f * w;  s[7] = cy + 0.5f * h;
        s[8] = w * h;
    }
    __syncthreads();

    const int wave   = tid >> 5;
    const int lane   = tid & 31;
    const int lane15 = lane & 15;
    const int hi     = lane >> 4;          // 0: lanes 0-15, 1: lanes 16-31
    const int koff   = hi ? 2 : 0;

    // this lane's target (one per lane, fixed across the 16x16 tile)
    const int t = blockIdx.x * 160 + wave * 16 + lane15;
    const float tcx = tgt_box[t * 4 + 0];
    const float tcy = tgt_box[t * 4 + 1];
    const float tw  = tgt_box[t * 4 + 2];
    const float th  = tgt_box[t * 4 + 3];
    const float tx0 = tcx - 0.5f * tw, ty0 = tcy - 0.5f * th;
    const float tx1 = tcx + 0.5f * tw, ty1 = tcy + 0.5f * th;
    const float tarea = tw * th;
    const int   tcls  = tgt_id[t];

    // ---- class cost via WMMA: Table(16x92) x OneHot(92x16), exact in f32 ----
    // A layout (ISA 7.12.2): lane<16 -> M=lane,K={4s,4s+1}; lane>=16 -> K={4s+2,4s+3}
    // B assumed mirrored: lane15 -> N, lane-half -> K parity pair.
    v8f acc = {0.f, 0.f, 0.f, 0.f, 0.f, 0.f, 0.f, 0.f};
    const int abase = lane15 * KPAD + koff;
#pragma unroll
    for (int s = 0; s < 23; ++s) {
        const int kk = 4 * s + koff;
        v2f av;
        av.x = ccs[abase + 4 * s];
        av.y = ccs[abase + 4 * s + 1];
        v2f bv;
        bv.x = (tcls == kk)     ? 1.0f : 0.0f;
        bv.y = (tcls == kk + 1) ? 1.0f : 0.0f;
        acc = __builtin_amdgcn_wmma_f32_16x16x4_f32(
                  false, av, false, bv, (short)0, acc, false, false);
    }

    // ---- combine with L1 + GIoU; D layout: VGPR r, lane -> (M=r+8*hi, N=lane15)
    const int hi8 = hi * 8;
#pragma unroll
    for (int r = 0; r < 8; ++r) {
        const int rl = r + hi8;
        const float* s = pbs + rl * 9;
        const float pcx = s[0], pcy = s[1], pw = s[2], ph = s[3];
        const float px0 = s[4], py0 = s[5], px1 = s[6], py1 = s[7];
        const float parea = s[8];

        const float l1 = fabsf(pcx - tcx) + fabsf(pcy - tcy)
                       + fabsf(pw  - tw ) + fabsf(ph  - th );

        const float ix0 = fmaxf(px0, tx0), iy0 = fmaxf(py0, ty0);
        const float ix1 = fminf(px1, tx1), iy1 = fminf(py1, ty1);
        const float iw = fmaxf(ix1 - ix0, 0.0f);
        const float ih = fmaxf(iy1 - iy0, 0.0f);
        const float inter = iw * ih;
        const float uni   = parea + tarea - inter;
        const float iou   = inter / uni;

        const float ex0 = fminf(px0, tx0), ey0 = fminf(py0, ty0);
        const float ex1 = fmaxf(px1, tx1), ey1 = fmaxf(py1, ty1);
        const float ew = fmaxf(ex1 - ex0, 0.0f);
        const float eh = fmaxf(ey1 - ey0, 0.0f);
        const float ear = ew * eh;
        const float giou = iou - (ear - uni) / ear;

        const float costv = acc[r] + COSTW_BBOX * l1 - COSTW_GIOU * giou;
        __builtin_nontemporal_store(costv, &out[(rowBase + rl) * NT + t]);
    }
}

// ---------------------------------------------------------------------------
extern "C" void kernel_launch(void* const* d_in, const int* in_sizes, int n_in,
                              void* d_out, int out_size, void* d_ws, size_t ws_size,
                              hipStream_t stream)
{
    const float* pred_logits = (const float*)d_in[0];  // [16,900,91]
    const float* pred_boxes  = (const float*)d_in[1];  // [16,900,4]
    const float* tgt_boxes   = (const float*)d_in[2];  // [1600,4]
    const int*   tgt_ids     = (const int*)  d_in[3];  // [1600]
    float*       out         = (float*)d_out;          // [16,900,1600]
    float*       cc          = (float*)d_ws;           // [14400][92] = 5.3 MB

    {
        int total  = NQ * KPAD;
        int blocks = (total + 255) / 256;
        hm_class_table<<<blocks, 256, 0, stream>>>(pred_logits, cc);
    }
    {
        dim3 grid(NT / 160, NQ / 16);   // (10, 900)
        hm_cost<<<grid, 320, 0, stream>>>(cc, pred_boxes, tgt_boxes, tgt_ids, out);
    }
}